// QKProjection_46806553591933
// MI455X (gfx1250) — compile-verified
//
#include <hip/hip_runtime.h>

// QKProjection: causal linear attention rewritten as GEMMs, fp32 WMMA (16x16x4).
//   out_b = tril(Q_b K_b^T) K_b + Q_b carry^T          (per batch)
//   new_carry = carry + (1/B) sum_b K_b^T K_b
// CDNA5 paths: v_wmma_f32_16x16x4_f32 + double-buffered global_load_async_to_lds.

typedef __attribute__((ext_vector_type(2))) float v2f;
typedef __attribute__((ext_vector_type(4))) float v4f;
typedef __attribute__((ext_vector_type(8))) float v8f;
typedef int v4i __attribute__((vector_size(16)));  // matches builtin pointee

typedef __attribute__((address_space(1))) v4i* gv4i_p;  // global int4*
typedef __attribute__((address_space(3))) v4i* lv4i_p;  // LDS int4*

#define WMMA_F32(a, b, c) \
  __builtin_amdgcn_wmma_f32_16x16x4_f32(false, (a), false, (b), (short)0, (c), false, false)

constexpr int Bn = 8;
constexpr int Cn = 512;
constexpr int Dn = 128;

#if __has_builtin(__builtin_amdgcn_global_load_async_to_lds_b128)
#define HAS_ASYNC_LDS 1
#else
#define HAS_ASYNC_LDS 0
#endif

// Stage one 16x128 fp32 K tile (8 KB) into LDS: 128 threads x 4 x 16B.
__device__ __forceinline__ void stage_ktile(const float* __restrict__ gsrc,
                                            float* lds, int tid) {
#if HAS_ASYNC_LDS
#pragma unroll
  for (int it = 0; it < 4; ++it) {
    const int idx = it * 512 + tid * 4;
    __builtin_amdgcn_global_load_async_to_lds_b128(
        (gv4i_p)(gsrc + idx), (lv4i_p)(lds + idx),
        /*offset=*/0, /*cpol=*/0);
  }
#else
#pragma unroll
  for (int it = 0; it < 4; ++it) {
    const int idx = it * 512 + tid * 4;
    *(v4f*)(lds + idx) = *(const v4f*)(gsrc + idx);
  }
#endif
}

__device__ __forceinline__ void wait_async0() {
#if HAS_ASYNC_LDS
#if __has_builtin(__builtin_amdgcn_s_wait_asynccnt)
  __builtin_amdgcn_s_wait_asynccnt(0);
#else
  asm volatile("s_wait_asynccnt 0x0" ::: "memory");
#endif
#endif
}

// ---------------------------------------------------------------------------
// Main kernel: 64 workgroups = 8 batches x 8 row-blocks (64 rows each).
// 128 threads = 4 waves; wave w owns rows [rb*64 + w*16, +16) x all 128 cols.
// ---------------------------------------------------------------------------
__global__ __launch_bounds__(128)
void qkproj_main(const float* __restrict__ Q, const float* __restrict__ K,
                 const float* __restrict__ carry, float* __restrict__ out)
{
  __shared__ __align__(16) float kBuf[2][16 * Dn];   // double-buffered K tiles
  __shared__ __align__(16) float pTile[4][16 * 16];  // per-wave P transpose buffer

  const int tid  = threadIdx.x;
  const int lane = tid & 31;
  const int wave = tid >> 5;
  const int lo   = lane & 15;   // 0..15
  const int hi   = lane >> 4;   // 0 or 1

  const int b       = blockIdx.x >> 3;  // batch
  const int rb      = blockIdx.x & 7;   // row block of 64
  const int rowbase = rb * 64;
  const int m0      = wave * 16;

  const float* Qb = Q + b * Cn * Dn;
  const float* Kb = K + b * Cn * Dn;

  // Prefetch K tile 0 asynchronously; it completes under the carry-term GEMM.
  stage_ktile(Kb, kBuf[0], tid);

  // --- preload Q A-fragments: A[M=lo][k = 4s + 2*hi + v], qrow fixed per lane
  const int qrow = rowbase + m0 + lo;
  v2f qa[32];
#pragma unroll
  for (int s = 0; s < 32; ++s)
    qa[s] = *(const v2f*)(Qb + qrow * Dn + s * 4 + 2 * hi);

  v8f acc[8];
#pragma unroll
  for (int t = 0; t < 8; ++t) acc[t] = (v8f){};

  // --- carry term: acc[t] += Q @ carry^T  (B[k=j][n] = carry[(t*16+n)*D + j])
#pragma unroll
  for (int t = 0; t < 8; ++t) {
    const float* cb = carry + (t * 16 + lo) * Dn + 2 * hi;
    v8f a = acc[t];
#pragma unroll
    for (int s = 0; s < 32; ++s) {
      v2f bf = *(const v2f*)(cb + s * 4);
      a = WMMA_F32(qa[s], bf, a);
    }
    acc[t] = a;
  }

  // --- causal attention: column tiles 0 .. rb*4+3 (uniform over block)
  const int numCb = rb * 4 + 4;
  for (int cb = 0; cb < numCb; ++cb) {
    const float* kT = kBuf[cb & 1];
    wait_async0();       // this wave's portion of current tile is in LDS
    __syncthreads();     // all waves' portions visible; prev buffer free
    if (cb + 1 < numCb)  // prefetch next tile into the other buffer
      stage_ktile(Kb + (cb + 1) * 16 * Dn, kBuf[(cb + 1) & 1], tid);

    if (cb <= rb * 4 + wave) {  // wave-uniform: block not fully above diagonal
      // P = Q @ Kcb^T : B[k=j][n=c'] = K[cb*16+n][j]
      v8f p = (v8f){};
#pragma unroll
      for (int s = 0; s < 32; ++s) {
        v2f bf = *(const v2f*)(kT + lo * Dn + s * 4 + 2 * hi);
        p = WMMA_F32(qa[s], bf, p);
      }
      // causal mask (inclusive) + store P to LDS in C/D layout:
      // element (M = v + 8*hi, N = lo)
      const int cglob = cb * 16 + lo;
      float* pw = pTile[wave];
#pragma unroll
      for (int v = 0; v < 8; ++v) {
        const int rglob = rowbase + m0 + v + 8 * hi;
        pw[(v + 8 * hi) * 16 + lo] = (cglob <= rglob) ? p[v] : 0.0f;
      }
      // acc[t] += P @ Kcb : K-dim = 16 in 4 steps of 4
#pragma unroll
      for (int kk = 0; kk < 16; kk += 4) {
        // A[M=lo][k = kk + 2*hi + v] from LDS (layout conversion)
        v2f af = *(const v2f*)(pw + lo * 16 + kk + 2 * hi);
#pragma unroll
        for (int t = 0; t < 8; ++t) {
          const int r0 = (kk + 2 * hi) * Dn + t * 16 + lo;
          v2f bf;
          bf.x = kT[r0];
          bf.y = kT[r0 + Dn];
          acc[t] = WMMA_F32(af, bf, acc[t]);
        }
      }
    }
  }

  // --- store: out[b][rowbase + m0 + v + 8*hi][t*16 + lo]
  float* ob = out + b * Cn * Dn;
#pragma unroll
  for (int t = 0; t < 8; ++t)
#pragma unroll
    for (int v = 0; v < 8; ++v)
      ob[(rowbase + m0 + v + 8 * hi) * Dn + t * 16 + lo] = acc[t][v];
}

// ---------------------------------------------------------------------------
// Carry kernel: new_carry = carry + (1/8) sum_{b,c} k (x) k.
// 64 workgroups (one 16x16 tile each), 4 waves split the 4096-long reduction;
// 2 interleaved accumulators per wave break the WMMA RAW chain.
// ---------------------------------------------------------------------------
__global__ __launch_bounds__(128)
void qkproj_carry(const float* __restrict__ K, const float* __restrict__ carry,
                  float* __restrict__ outc)
{
  __shared__ float red[4][32 * 8];

  const int tid  = threadIdx.x;
  const int lane = tid & 31;
  const int wave = tid >> 5;
  const int lo   = lane & 15;
  const int hi   = lane >> 4;
  const int ti   = blockIdx.x >> 3;  // row tile of result
  const int tj   = blockIdx.x & 7;   // col tile of result

  const int ca  = ti * 16 + lo;  // column of K feeding A (result row index)
  const int cbn = tj * 16 + lo;  // column of K feeding B (result col index)

  v8f s0 = (v8f){}, s1 = (v8f){};
  const int c_begin = wave * (Bn * Cn / 4);
  const int c_end   = c_begin + (Bn * Cn / 4);
  for (int c0 = c_begin; c0 < c_end; c0 += 8) {
    const int r0 = c0 + 2 * hi;
    const int r1 = c0 + 4 + 2 * hi;
    v2f a0, b0, a1, b1;
    a0.x = K[r0 * Dn + ca];        a0.y = K[(r0 + 1) * Dn + ca];
    b0.x = K[r0 * Dn + cbn];       b0.y = K[(r0 + 1) * Dn + cbn];
    a1.x = K[r1 * Dn + ca];        a1.y = K[(r1 + 1) * Dn + ca];
    b1.x = K[r1 * Dn + cbn];       b1.y = K[(r1 + 1) * Dn + cbn];
    s0 = WMMA_F32(a0, b0, s0);
    s1 = WMMA_F32(a1, b1, s1);
  }
  v8f s = s0 + s1;

#pragma unroll
  for (int v = 0; v < 8; ++v) red[wave][lane * 8 + v] = s[v];
  __syncthreads();

  if (wave == 0) {
#pragma unroll
    for (int v = 0; v < 8; ++v) {
      const float sum = red[0][lane * 8 + v] + red[1][lane * 8 + v] +
                        red[2][lane * 8 + v] + red[3][lane * 8 + v];
      const int i = ti * 16 + v + 8 * hi;
      const int j = tj * 16 + lo;
      outc[i * Dn + j] = carry[i * Dn + j] + 0.125f * sum;
    }
  }
}

// ---------------------------------------------------------------------------
extern "C" void kernel_launch(void* const* d_in, const int* in_sizes, int n_in,
                              void* d_out, int out_size, void* d_ws, size_t ws_size,
                              hipStream_t stream)
{
  const float* Q     = (const float*)d_in[0];  // queries  (8,512,128)
  const float* K     = (const float*)d_in[1];  // keys     (8,512,128)
  const float* carry = (const float*)d_in[2];  // carry    (128,128)
  float* out  = (float*)d_out;                 // projected_queries flat
  float* outc = out + Bn * Cn * Dn;            // new_carry flat

  qkproj_main<<<dim3(Bn * (Cn / 64)), dim3(128), 0, stream>>>(Q, K, carry, out);
  qkproj_carry<<<dim3(64), dim3(128), 0, stream>>>(K, carry, outc);
}